// Hopf_20495583936582
// MI455X (gfx1250) — compile-verified
//
#include <hip/hip_runtime.h>
#include <cstdint>

#define DT_C           0.01f
#define INPUT_SCALER_C 20.0f

// ---------------------------------------------------------------------------
// CDNA5 async global -> LDS copy (ASYNCcnt-tracked), GVS addressing mode:
//   LDS destination: per-lane LDS byte address in vdst
//   global source: 64-bit SGPR base + 32-bit VGPR byte offset
// ---------------------------------------------------------------------------
__device__ __forceinline__ void async_ld_b32(uint32_t lds_addr, uint32_t goff,
                                             const void* base) {
  asm volatile("global_load_async_to_lds_b32 %0, %1, %2"
               :: "v"(lds_addr), "v"(goff), "s"(base)
               : "memory");
}

// Generic pointer to __shared__ has the workgroup-relative LDS byte offset in
// its low 32 bits (ISA 10.2).
__device__ __forceinline__ uint32_t lds_off(const void* p) {
  return (uint32_t)(uintptr_t)p;
}

// One thread per oscillator chain; scan over T=64 timesteps.
// Layout [bs, T, d, d, nk]: elem(b,t,inner) = (b<<21) + (t<<15) + inner,
// with inner = d*d*nk index in [0, 32768).
__global__ __launch_bounds__(256) void hopf_scan_kernel(
    const float* __restrict__ Xr, const float* __restrict__ Xi,
    const float* __restrict__ omegas,
    float* __restrict__ out_r, float* __restrict__ out_i) {
  __shared__ float sh_xr[4][256];   // 4-deep pipeline buffers, one slot/lane
  __shared__ float sh_xi[4][256];

  const int      tid       = threadIdx.x;
  const uint32_t chain     = blockIdx.x * 256u + (uint32_t)tid;
  const uint32_t b         = chain >> 15;        // / 32768
  const uint32_t inner     = chain & 32767u;     // % 32768
  const uint32_t base_elem = (b << 21) + inner;  // b * T * 32768 + inner

  const float omega = omegas[inner];             // 128 KiB, L2-resident

  uint32_t a_xr[4], a_xi[4];
#pragma unroll
  for (int i = 0; i < 4; ++i) {
    a_xr[i] = lds_off(&sh_xr[i][tid]);
    a_xi[i] = lds_off(&sh_xi[i][tid]);
  }

  // Prologue: stream timesteps 0..2 into buffers 0..2 (6 async ops in flight).
#pragma unroll
  for (int p = 0; p < 3; ++p) {
    const uint32_t goff = (base_elem + ((uint32_t)p << 15)) << 2;
    async_ld_b32(a_xr[p], goff, Xr);
    async_ld_b32(a_xi[p], goff, Xi);
  }

  float r = 1.0f, phi = 0.0f;
  float cphi = 1.0f, sphi = 0.0f;  // carried cos/sin(phi): step t's output
                                   // pair doubles as step t+1's input pair.

  for (int t = 0; t < 64; ++t) {
    const int buf = t & 3;

    if (t + 3 < 64) {
      // Keep 4 pairs in flight; async loads retire in order, so ASYNCcnt<=6
      // after issuing (t+3) guarantees timestep t's pair is in LDS.
      const uint32_t goff = (base_elem + ((uint32_t)(t + 3) << 15)) << 2;
      async_ld_b32(a_xr[(t + 3) & 3], goff, Xr);
      async_ld_b32(a_xi[(t + 3) & 3], goff, Xi);
      asm volatile("s_wait_asynccnt 0x6" ::: "memory");
    } else {
      asm volatile("s_wait_asynccnt 0x0" ::: "memory");
    }

    const float xr = sh_xr[buf][tid];
    const float xi = sh_xi[buf][tid];

    const float in_r = INPUT_SCALER_C * xr * cphi;
    const float in_p = INPUT_SCALER_C * xi * sphi;
    r   = r + ((1.0f - r * r) * r + in_r) * DT_C;
    phi = phi + (omega - in_p) * DT_C;

    cphi = __cosf(phi);   // v_cos_f32 (hardware transcendental)
    sphi = __sinf(phi);   // v_sin_f32

    const uint32_t e = base_elem + ((uint32_t)t << 15);
    out_r[e] = r * cphi;
    out_i[e] = r * sphi;
  }
}

extern "C" void kernel_launch(void* const* d_in, const int* in_sizes, int n_in,
                              void* d_out, int out_size, void* d_ws, size_t ws_size,
                              hipStream_t stream) {
  (void)n_in; (void)d_ws; (void)ws_size; (void)out_size;
  const float* Xr = (const float*)d_in[0];
  const float* Xi = (const float*)d_in[1];
  const float* om = (const float*)d_in[2];

  const int total  = in_sizes[0];     // bs*T*d*d*nk = 16,777,216
  const int chains = total / 64;      // 262,144 independent chains

  float* out_r = (float*)d_out;
  float* out_i = out_r + total;       // tuple output: (real, imag) concatenated

  dim3 block(256);
  dim3 grid((chains + 255) / 256);    // 1024 blocks = 8192 wave32 waves
  hopf_scan_kernel<<<grid, block, 0, stream>>>(Xr, Xi, om, out_r, out_i);
}